// GraphLayerNormImproved_72224170049726
// MI455X (gfx1250) — compile-verified
//
#include <hip/hip_runtime.h>
#include <math.h>

#define GLN_C     256     // channels
#define GLN_MAXN  128     // max nodes per graph staged in LDS (65536/512 = 128)
#define GLN_TPB   256     // 8 wave32s
#define GLN_NW    8
#define GLN_EPS   1e-6f

struct GlnSmem {
  float ss[GLN_MAXN * GLN_C];     // staged s tile, 128 KB (async DMA target)
  float colred[GLN_NW * GLN_C];   // per-wave column-sum partials, 8 KB
  float buf[GLN_TPB];             // block reduction scratch
  int   ibuf[GLN_TPB];
  float rowmean[GLN_MAXN];
  float meang[GLN_C];
  float wsh[GLN_C];
  float bsh[GLN_C];
};

__device__ __forceinline__ float gln_wave_sum(float x) {
#pragma unroll
  for (int i = 16; i > 0; i >>= 1) x += __shfl_xor(x, i, 32);
  return x;
}

__device__ __forceinline__ float gln_block_sum(float x, float* buf) {
  const int tid = threadIdx.x;
  buf[tid] = x;
  __syncthreads();
#pragma unroll
  for (int s = GLN_TPB >> 1; s > 0; s >>= 1) {
    if (tid < s) buf[tid] += buf[tid + s];
    __syncthreads();
  }
  float r = buf[0];
  __syncthreads();
  return r;
}

__device__ __forceinline__ int gln_block_sum_i(int x, int* buf) {
  const int tid = threadIdx.x;
  buf[tid] = x;
  __syncthreads();
#pragma unroll
  for (int s = GLN_TPB >> 1; s > 0; s >>= 1) {
    if (tid < s) buf[tid] += buf[tid + s];
    __syncthreads();
  }
  int r = buf[0];
  __syncthreads();
  return r;
}

// native clang vector for non-temporal builtins (HIP float4 is a class type)
typedef float gln_v4 __attribute__((ext_vector_type(4)));

// last-use (non-temporal) float4 load / stream (non-temporal) float4 store
__device__ __forceinline__ float4 gln_ld_nt(const float* p) {
  gln_v4 t = __builtin_nontemporal_load((const gln_v4*)p);
  return make_float4(t.x, t.y, t.z, t.w);
}
__device__ __forceinline__ void gln_st_nt(float* p, float4 x) {
  gln_v4 t;
  t.x = x.x; t.y = x.y; t.z = x.z; t.w = x.w;
  __builtin_nontemporal_store(t, (gln_v4*)p);
}

__global__ __launch_bounds__(GLN_TPB, 1)
void gln_kernel(const float* __restrict__ s, const float* __restrict__ v,
                const float* __restrict__ weight, const float* __restrict__ bias,
                const int* __restrict__ splits,
                float* __restrict__ sout, float* __restrict__ vout) {
  __shared__ GlnSmem sm;
  const int tid  = threadIdx.x;
  const int g    = blockIdx.x;
  const int lane = tid & 31;
  const int wv   = tid >> 5;
  const int c0   = lane * 4;          // channels [4l, 4l+3]
  const int c1   = 128 + lane * 4;    // channels [128+4l, 128+4l+3]

  // weight/bias -> LDS (tid == channel, TPB == C)
  sm.wsh[tid] = weight[tid];
  sm.bsh[tid] = bias[tid];

  // node offset of this graph = prefix sum of splits[0:g)
  int acc = 0;
  for (int i = tid; i < g; i += GLN_TPB) acc += splits[i];
  const int start = gln_block_sum_i(acc, sm.ibuf);
  int cnt = splits[g];
  if (cnt > GLN_MAXN) cnt = GLN_MAXN;

  // ---- CDNA5 async DMA: stage s tile (cnt x 256 f32) global -> LDS, no VGPRs
  // GV addressing: 64-bit address in a VGPR pair, saddr = off
  {
    const unsigned long long gbase =
        (unsigned long long)(const void*)(s + (size_t)start * GLN_C);
    const unsigned ldsbase = (unsigned)(unsigned long long)(void*)&sm.ss[0];
    const int chunks = cnt * (GLN_C / 4);       // float4 chunks
    for (int i = tid; i < chunks; i += GLN_TPB) {
      unsigned long long gaddr = gbase + (unsigned long long)((unsigned)i * 16u);
      unsigned laddr = ldsbase + (unsigned)i * 16u;
      asm volatile("global_load_async_to_lds_b128 %0, %1, off"
                   :: "v"(laddr), "v"(gaddr) : "memory");
    }
  }

  // ---- vector branch pass 1: sum over (n,c) of sqrt(v0^2+v1^2+v2^2+eps)
  // (overlaps with the async DMA above; regular-temporal loads so the lines
  //  stay L2-resident for pass 2)
  float vpart = 0.f;
  for (int r = wv; r < cnt; r += GLN_NW) {
    const float* vb = v + (size_t)(start + r) * (3 * GLN_C);
    float4 x0 = *(const float4*)(vb + c0);
    float4 x1 = *(const float4*)(vb + c1);
    float4 y0 = *(const float4*)(vb + GLN_C + c0);
    float4 y1 = *(const float4*)(vb + GLN_C + c1);
    float4 z0 = *(const float4*)(vb + 2 * GLN_C + c0);
    float4 z1 = *(const float4*)(vb + 2 * GLN_C + c1);
    vpart += sqrtf(x0.x*x0.x + y0.x*y0.x + z0.x*z0.x + GLN_EPS)
           + sqrtf(x0.y*x0.y + y0.y*y0.y + z0.y*z0.y + GLN_EPS)
           + sqrtf(x0.z*x0.z + y0.z*y0.z + z0.z*z0.z + GLN_EPS)
           + sqrtf(x0.w*x0.w + y0.w*y0.w + z0.w*z0.w + GLN_EPS)
           + sqrtf(x1.x*x1.x + y1.x*y1.x + z1.x*z1.x + GLN_EPS)
           + sqrtf(x1.y*x1.y + y1.y*y1.y + z1.y*z1.y + GLN_EPS)
           + sqrtf(x1.z*x1.z + y1.z*y1.z + z1.z*z1.z + GLN_EPS)
           + sqrtf(x1.w*x1.w + y1.w*y1.w + z1.w*z1.w + GLN_EPS);
  }
  const float vsum    = gln_block_sum(vpart, sm.buf);
  const float vnorm_g = vsum * (1.0f / ((float)cnt * (float)GLN_C));
  const float vinv    = 1.0f / vnorm_g;

  // ---- vector branch pass 2: vout = v / vnorm_g
  // v re-read is last-use (NT load, still hits L2); vout is a pure write
  // stream (NT store, don't pollute L2)
  for (int r = wv; r < cnt; r += GLN_NW) {
    const float* vb = v    + (size_t)(start + r) * (3 * GLN_C);
    float*       vo = vout + (size_t)(start + r) * (3 * GLN_C);
#pragma unroll
    for (int p = 0; p < 3; ++p) {
      float4 a = gln_ld_nt(vb + p * GLN_C + c0);
      float4 b = gln_ld_nt(vb + p * GLN_C + c1);
      a.x *= vinv; a.y *= vinv; a.z *= vinv; a.w *= vinv;
      b.x *= vinv; b.y *= vinv; b.z *= vinv; b.w *= vinv;
      gln_st_nt(vo + p * GLN_C + c0, a);
      gln_st_nt(vo + p * GLN_C + c1, b);
    }
  }

  // ---- wait for staged s tile (ASYNCcnt), then make it visible to all waves
  asm volatile("s_wait_asynccnt 0" ::: "memory");
  __syncthreads();

  // ---- s pass 1 (LDS): per-row means + per-wave column sums
  float4 ca0 = make_float4(0.f, 0.f, 0.f, 0.f);
  float4 ca1 = make_float4(0.f, 0.f, 0.f, 0.f);
  for (int r = wv; r < cnt; r += GLN_NW) {
    float4 x0 = *(const float4*)&sm.ss[r * GLN_C + c0];
    float4 x1 = *(const float4*)&sm.ss[r * GLN_C + c1];
    float rp = x0.x + x0.y + x0.z + x0.w + x1.x + x1.y + x1.z + x1.w;
    float rs = gln_wave_sum(rp);
    if (lane == 0) sm.rowmean[r] = rs * (1.0f / GLN_C);
    ca0.x += x0.x; ca0.y += x0.y; ca0.z += x0.z; ca0.w += x0.w;
    ca1.x += x1.x; ca1.y += x1.y; ca1.z += x1.z; ca1.w += x1.w;
  }
  *(float4*)&sm.colred[wv * GLN_C + c0] = ca0;
  *(float4*)&sm.colred[wv * GLN_C + c1] = ca1;
  __syncthreads();

  // mean_g[c] = (colsum[c] - sum_n rowmean[n]) / cnt
  const float rmsum = gln_block_sum(tid < cnt ? sm.rowmean[tid] : 0.f, sm.buf);
  float csum = 0.f;
#pragma unroll
  for (int q = 0; q < GLN_NW; ++q) csum += sm.colred[q * GLN_C + tid];
  sm.meang[tid] = (csum - rmsum) / (float)cnt;
  __syncthreads();

  const float4 m0 = *(const float4*)&sm.meang[c0];
  const float4 m1 = *(const float4*)&sm.meang[c1];

  // ---- s pass 2 (LDS): variance = mean over (n,c) of cs^2
  float ssq = 0.f;
  for (int r = wv; r < cnt; r += GLN_NW) {
    float rm = sm.rowmean[r];
    float4 x0 = *(const float4*)&sm.ss[r * GLN_C + c0];
    float4 x1 = *(const float4*)&sm.ss[r * GLN_C + c1];
    float d;
    d = x0.x - rm - m0.x; ssq += d * d;
    d = x0.y - rm - m0.y; ssq += d * d;
    d = x0.z - rm - m0.z; ssq += d * d;
    d = x0.w - rm - m0.w; ssq += d * d;
    d = x1.x - rm - m1.x; ssq += d * d;
    d = x1.y - rm - m1.y; ssq += d * d;
    d = x1.z - rm - m1.z; ssq += d * d;
    d = x1.w - rm - m1.w; ssq += d * d;
  }
  const float var_g = gln_block_sum(ssq, sm.buf) *
                      (1.0f / ((float)cnt * (float)GLN_C));
  const float istd = 1.0f / sqrtf(var_g + GLN_EPS);

  // ---- s pass 3: sout = cs * istd * weight + bias (tile from LDS, NT stores)
  const float4 w0 = *(const float4*)&sm.wsh[c0];
  const float4 w1 = *(const float4*)&sm.wsh[c1];
  const float4 b0 = *(const float4*)&sm.bsh[c0];
  const float4 b1 = *(const float4*)&sm.bsh[c1];
  for (int r = wv; r < cnt; r += GLN_NW) {
    float rm = sm.rowmean[r];
    float4 x0 = *(const float4*)&sm.ss[r * GLN_C + c0];
    float4 x1 = *(const float4*)&sm.ss[r * GLN_C + c1];
    float4 o0, o1;
    o0.x = (x0.x - rm - m0.x) * istd * w0.x + b0.x;
    o0.y = (x0.y - rm - m0.y) * istd * w0.y + b0.y;
    o0.z = (x0.z - rm - m0.z) * istd * w0.z + b0.z;
    o0.w = (x0.w - rm - m0.w) * istd * w0.w + b0.w;
    o1.x = (x1.x - rm - m1.x) * istd * w1.x + b1.x;
    o1.y = (x1.y - rm - m1.y) * istd * w1.y + b1.y;
    o1.z = (x1.z - rm - m1.z) * istd * w1.z + b1.z;
    o1.w = (x1.w - rm - m1.w) * istd * w1.w + b1.w;
    float* so = sout + (size_t)(start + r) * GLN_C;
    gln_st_nt(so + c0, o0);
    gln_st_nt(so + c1, o1);
  }
}

extern "C" void kernel_launch(void* const* d_in, const int* in_sizes, int n_in,
                              void* d_out, int out_size, void* d_ws, size_t ws_size,
                              hipStream_t stream) {
  (void)n_in; (void)out_size; (void)d_ws; (void)ws_size;
  const float* s      = (const float*)d_in[0];
  const float* v      = (const float*)d_in[1];
  const float* weight = (const float*)d_in[2];
  const float* bias   = (const float*)d_in[3];
  const int*   splits = (const int*)d_in[4];
  const int G = in_sizes[4];
  const size_t N = (size_t)in_sizes[0] / GLN_C;
  float* sout = (float*)d_out;
  float* vout = sout + N * GLN_C;
  gln_kernel<<<G, GLN_TPB, 0, stream>>>(s, v, weight, bias, splits, sout, vout);
}